// SimpleVQSign_71107478553202
// MI455X (gfx1250) — compile-verified
//
#include <hip/hip_runtime.h>
#include <hip/hip_bf16.h>

// CDNA5 / gfx1250 fp32 WMMA: D(16x16,f32) = A(16x4,f32) x B(4x16,f32) + C
// A/B operands: 64 fp32 over 32 lanes -> float2 per lane. C/D: float8 per lane.
typedef __attribute__((ext_vector_type(2))) float v2f;
typedef __attribute__((ext_vector_type(8))) float v8f;

__device__ __forceinline__ v8f wmma_f32_4(v2f a, v2f b, v8f c) {
    // 8 args: (neg_a, A, neg_b, B, c_mod, C, reuse_a, reuse_b)
    return __builtin_amdgcn_wmma_f32_16x16x4_f32(false, a, false, b, (short)0, c,
                                                 false, false);
}

// ---------------------------------------------------------------------------
// Tiled fp32 WMMA GEMM:  C[M,N] = op( A[M,K] * B + bias )
//   TRANS_B=0: B is [K,N] row-major (ldb = N)
//   TRANS_B=1: B is [N,K] row-major (ldb = K), i.e. C = A * B^T
//
// Block: 256 threads (8 waves). Output block 128(M) x 128(N).
// Wave (wm = wave&3, wn = wave>>2): rows [wm*32, +32) x cols [wn*64, +64)
//   -> 2x4 = 8 accumulator tiles of 16x16 per wave.
// K staged in double-buffered LDS chunks of KC=16:
//   As[buf][128][18]      row-major, padded (even stride keeps b64 align,
//                         18-float stride spreads banks)
//   Bs[buf][8][128][2]    K-PAIRED: element (k,n) at [k>>1][n][k&1] so a
//                         B fragment (K=kk+2h, kk+2h+1) is ONE aligned b64.
// ---------------------------------------------------------------------------
template <bool TRANS_B, bool RELU, bool HAS_BIAS>
__global__ void __launch_bounds__(256)
gemm_wmma_f32(const float* __restrict__ A, const float* __restrict__ B,
              const float* __restrict__ bias, float* __restrict__ C,
              int K, int lda, int ldb, int ldc)
{
    __shared__ __align__(16) float As[2][128][18];
    __shared__ __align__(16) float Bs[2][8][128][2];

    const int tid  = threadIdx.x;
    const int lane = tid & 31;
    const int wave = tid >> 5;
    const int wm   = wave & 3;                 // 32-row strip
    const int wn   = wave >> 2;                // 64-col strip (0..1)
    const int half = (lane >> 4) & 1;          // lanes 16-31 carry K+2/K+3
    const int l15  = lane & 15;
    const int rowBase = blockIdx.y * 128;
    const int colBase = blockIdx.x * 128;

    // staging slot decomposition (per thread: 2 float4 for A, 2 for B)
    const int ar0 = tid >> 2, ac4 = (tid & 3) << 2;          // A: 128 rows x 4 float4
    const int bn0 = tid >> 2, bk4 = (tid & 3) << 2;          // B^T: 128 n x 4 k-float4
    const int br0 = tid >> 5, bc4 = (tid & 31) << 2;         // B:   16 k x 32 n-float4

    v8f acc[2][4] = {};

    const int nch = K >> 4;                    // KC = 16
    float4 ra[2], rb[2];

    // ---- helpers (inlined manually) ----
    auto load_chunk = [&](int ch) {
        const int k0 = ch << 4;
#pragma unroll
        for (int i = 0; i < 2; ++i)
            ra[i] = *(const float4*)(A + (size_t)(rowBase + ar0 + i * 64) * lda + k0 + ac4);
        if (!TRANS_B) {
#pragma unroll
            for (int i = 0; i < 2; ++i)
                rb[i] = *(const float4*)(B + (size_t)(k0 + br0 + i * 8) * ldb + colBase + bc4);
        } else {
#pragma unroll
            for (int i = 0; i < 2; ++i)
                rb[i] = *(const float4*)(B + (size_t)(colBase + bn0 + i * 64) * ldb + k0 + bk4);
        }
    };
    auto store_chunk = [&](int buf) {
#pragma unroll
        for (int i = 0; i < 2; ++i) {
            As[buf][ar0 + i * 64][ac4 + 0] = ra[i].x;
            As[buf][ar0 + i * 64][ac4 + 1] = ra[i].y;
            As[buf][ar0 + i * 64][ac4 + 2] = ra[i].z;
            As[buf][ar0 + i * 64][ac4 + 3] = ra[i].w;
        }
        if (!TRANS_B) {
#pragma unroll
            for (int i = 0; i < 2; ++i) {
                const int k = br0 + i * 8;
                Bs[buf][k >> 1][bc4 + 0][k & 1] = rb[i].x;
                Bs[buf][k >> 1][bc4 + 1][k & 1] = rb[i].y;
                Bs[buf][k >> 1][bc4 + 2][k & 1] = rb[i].z;
                Bs[buf][k >> 1][bc4 + 3][k & 1] = rb[i].w;
            }
        } else {
#pragma unroll
            for (int i = 0; i < 2; ++i) {
                const int n = bn0 + i * 64, p = bk4 >> 1;
                v2f lo; lo.x = rb[i].x; lo.y = rb[i].y;
                v2f hi; hi.x = rb[i].z; hi.y = rb[i].w;
                *(v2f*)&Bs[buf][p + 0][n][0] = lo;
                *(v2f*)&Bs[buf][p + 1][n][0] = hi;
            }
        }
    };

    // ---- software pipeline: global->regs (i+1) | compute (i) | regs->LDS ----
    load_chunk(0);
    store_chunk(0);
    __syncthreads();

    int buf = 0;
    for (int ch = 0; ch < nch; ++ch) {
        const bool more = (ch + 1 < nch);
        if (more) load_chunk(ch + 1);

#pragma unroll
        for (int kk = 0; kk < 16; kk += 4) {
            v2f a0 = *(const v2f*)&As[buf][wm * 32 + l15][kk + 2 * half];
            v2f a1 = *(const v2f*)&As[buf][wm * 32 + 16 + l15][kk + 2 * half];
#pragma unroll
            for (int t = 0; t < 4; ++t) {
                const int bc = wn * 64 + t * 16 + l15;
                const v2f b = *(const v2f*)&Bs[buf][(kk >> 1) + half][bc][0];
                acc[0][t] = wmma_f32_4(a0, b, acc[0][t]);
                acc[1][t] = wmma_f32_4(a1, b, acc[1][t]);
            }
        }

        if (more) {
            store_chunk(buf ^ 1);
            __syncthreads();
            buf ^= 1;
        }
    }

    // ---- epilogue: bias + relu + store ----
#pragma unroll
    for (int t = 0; t < 4; ++t) {
        const int col = colBase + wn * 64 + t * 16 + l15;
        const float bv = HAS_BIAS ? bias[col] : 0.0f;
#pragma unroll
        for (int sub = 0; sub < 2; ++sub) {
#pragma unroll
            for (int r = 0; r < 8; ++r) {
                const int row = rowBase + wm * 32 + sub * 16 + r + half * 8;
                float v = acc[sub][t][r] + bv;
                if (RELU) v = fmaxf(v, 0.0f);
                C[(size_t)row * ldc + col] = v;
            }
        }
    }
}

// ---------------------------------------------------------------------------
// codebook norms (256 x 512) + zero the loss accumulator
// ---------------------------------------------------------------------------
__global__ void cbnorm_kernel(const float* __restrict__ cb,
                              float* __restrict__ cbn, float* __restrict__ acc)
{
    const int k = threadIdx.x;                 // 256 threads, one code each
    float s = 0.0f;
    for (int d = 0; d < 512; ++d) {
        const float v = cb[(size_t)k * 512 + d];
        s += v * v;
    }
    cbn[k] = s;
    if (k == 0) *acc = 0.0f;
}

// ---------------------------------------------------------------------------
// Per-row: argmin_k (||cb_k||^2 - 2 s_k), emit index (as f32), gather
// quantized row, accumulate ||e||^2 + min-value (== ||e - cb_sel||^2).
// One wave (32 lanes) per row; 8 rows per 256-thread block.
// ---------------------------------------------------------------------------
__global__ void __launch_bounds__(256)
argmin_quant_kernel(const float* __restrict__ enc, const float* __restrict__ scores,
                    const float* __restrict__ cbn, const float* __restrict__ cb,
                    float* __restrict__ out_idx, float* __restrict__ out_q,
                    float* __restrict__ acc)
{
    const int lane = threadIdx.x & 31;
    const int wave = threadIdx.x >> 5;
    const int row  = blockIdx.x * 8 + wave;

    // argmin over 256 codes (first-min tie-break, matching jnp.argmin)
    float bestv = 3.402823466e38f;
    int   besti = 0;
#pragma unroll
    for (int i = 0; i < 8; ++i) {
        const int k = lane + i * 32;
        const float d = cbn[k] - 2.0f * scores[(size_t)row * 256 + k];
        if (d < bestv) { bestv = d; besti = k; }
    }
#pragma unroll
    for (int off = 16; off > 0; off >>= 1) {
        const float ov = __shfl_down(bestv, off, 32);
        const int   oi = __shfl_down(besti, off, 32);
        if (ov < bestv || (ov == bestv && oi < besti)) { bestv = ov; besti = oi; }
    }
    besti = __shfl(besti, 0, 32);
    bestv = __shfl(bestv, 0, 32);

    // ||e||^2
    float e2 = 0.0f;
#pragma unroll
    for (int i = 0; i < 16; ++i) {
        const float e = enc[(size_t)row * 512 + lane + i * 32];
        e2 += e * e;
    }
#pragma unroll
    for (int off = 16; off > 0; off >>= 1) e2 += __shfl_down(e2, off, 32);

    // quantized row = codebook[besti]
#pragma unroll
    for (int i = 0; i < 16; ++i) {
        const int d = lane + i * 32;
        out_q[(size_t)row * 512 + d] = cb[(size_t)besti * 512 + d];
    }

    if (lane == 0) {
        out_idx[row] = (float)besti;
        atomicAdd(acc, e2 + bestv);            // ||e||^2 - 2 s_sel + ||cb_sel||^2
    }
}

__global__ void finalize_kernel(const float* __restrict__ acc, float* __restrict__ out_s)
{
    const float mean = *acc / (32768.0f * 512.0f);
    out_s[0] = mean;            // commitment_loss
    out_s[1] = mean;            // codebook_loss (same forward value)
    out_s[2] = 1.25f * mean;    // vq_loss
}

// ---------------------------------------------------------------------------
// inputs: x[64,512,1024], W1[1024,512], b1[512], W2[512,512], b2[512], cb[256,512]
// d_out : idx (32768 f32) | quantized (32768*512 f32) | 3 scalars
// d_ws  : [0,64MB) h (later aliased by scores) | [64MB,128MB) enc |
//         [128MB,+1KB) cbnorms | +4B loss accumulator   (needs >= ~128 MB + 2 KB)
// ---------------------------------------------------------------------------
extern "C" void kernel_launch(void* const* d_in, const int* in_sizes, int n_in,
                              void* d_out, int out_size, void* d_ws, size_t ws_size,
                              hipStream_t stream)
{
    (void)in_sizes; (void)n_in; (void)out_size; (void)ws_size;
    const float* x  = (const float*)d_in[0];
    const float* W1 = (const float*)d_in[1];
    const float* b1 = (const float*)d_in[2];
    const float* W2 = (const float*)d_in[3];
    const float* b2 = (const float*)d_in[4];
    const float* cb = (const float*)d_in[5];

    const int M = 64 * 512;                                   // 32768 rows

    float* h      = (float*)d_ws;                             // 64 MB
    float* enc    = (float*)((char*)d_ws + ((size_t)64 << 20));
    float* scores = (float*)d_ws;                             // aliases h (dead after GEMM2)
    float* cbn    = (float*)((char*)d_ws + ((size_t)128 << 20));
    float* acc    = cbn + 256;

    float* out_idx = (float*)d_out;
    float* out_q   = out_idx + M;
    float* out_s   = out_q + (size_t)M * 512;

    cbnorm_kernel<<<1, 256, 0, stream>>>(cb, cbn, acc);

    // GEMM1: h = relu(x @ W1 + b1)   [32768 x 1024] x [1024 x 512]
    gemm_wmma_f32<false, true, true>
        <<<dim3(512 / 128, M / 128), 256, 0, stream>>>(x, W1, b1, h, 1024, 1024, 512, 512);

    // GEMM2: enc = h @ W2 + b2       [32768 x 512] x [512 x 512]
    gemm_wmma_f32<false, false, true>
        <<<dim3(512 / 128, M / 128), 256, 0, stream>>>(h, W2, b2, enc, 512, 512, 512, 512);

    // GEMM3: scores = enc @ cb^T     [32768 x 512] x [512 x 256]^T
    gemm_wmma_f32<true, false, false>
        <<<dim3(256 / 128, M / 128), 256, 0, stream>>>(enc, cb, nullptr, scores, 512, 512, 512, 256);

    argmin_quant_kernel<<<M / 8, 256, 0, stream>>>(enc, scores, cbn, cb, out_idx, out_q, acc);
    finalize_kernel<<<1, 1, 0, stream>>>(acc, out_s);
}